// LennardJonesPurePyTorch_89361089560859
// MI455X (gfx1250) — compile-verified
//
#include <hip/hip_runtime.h>
#include <stdint.h>

// ---------------------------------------------------------------------------
// Lennard-Jones half-list pair kernel for MI455X (gfx1250).
//  - streams `distances` through LDS with gfx1250 async global->LDS b128
//    loads (ASYNCcnt, per-wave double buffer, s_wait_asynccnt pipeline),
//    non-temporal so the 168 MB read-once stream doesn't evict the 4 MB
//    atomic-accumulation region from the 192 MB L2
//  - 128 pairs staged per wave per stage (3 x b128 per lane), 4 pairs
//    computed per lane per stage -> 4x less pipeline/loop overhead per pair
//  - per-pair math in f32 (r^-12 dynamic range requires it)
//  - scatter with no-return global_atomic_add_f32 (output lives in L2)
// ---------------------------------------------------------------------------

namespace {
constexpr float EPSILON = 1.0f;
constexpr float SIGMA   = 1.0f;
constexpr float CUTOFF  = 5.0f;
constexpr float SRC2    = (SIGMA * SIGMA) / (CUTOFF * CUTOFF);
constexpr float SR6C    = SRC2 * SRC2 * SRC2;
constexpr float LJSHIFT = 4.0f * EPSILON * (SR6C * SR6C - SR6C);

constexpr int TILE            = 128;            // pairs per wave per stage
constexpr int SUB             = TILE / 32;      // pairs per lane per stage (4)
constexpr int TILE_BYTES      = TILE * 12;      // 1536 B of distances
constexpr int WAVES_PER_BLOCK = 8;              // 256 threads, wave32
constexpr int TILES_PER_WAVE  = 8;              // target stages per wave
}  // namespace

typedef const __attribute__((address_space(3))) void lds_const_void;

__device__ __forceinline__ uint32_t lds_byte_off(const void* p) {
  // generic -> LDS addrspace cast; ptrtoint of an AS(3) pointer is the
  // 32-bit LDS byte offset the async-load VDST operand wants.
  return (uint32_t)(size_t)(lds_const_void*)p;
}

__device__ __forceinline__ void lj_pair(float dx, float dy, float dz,
                                        int ai, int aj,
                                        float* __restrict__ out) {
  const float r2   = dx * dx + dy * dy + dz * dz;
  const float inv  = 1.0f / r2;
  const float sr2  = (SIGMA * SIGMA) * inv;
  const float sr6  = sr2 * sr2 * sr2;
  const float sr12 = sr6 * sr6;
  const float e    = 4.0f * EPSILON * (sr12 - sr6) - LJSHIFT;
  const float fs   = 12.0f * EPSILON * (sr6 - 2.0f * sr12) * inv;
  const float he   = 0.5f * e;
  const float fx = fs * dx, fy = fs * dy, fz = fs * dz;
  float* oi = out + 4 * ai;
  float* oj = out + 4 * aj;
  // no-return f32 adds -> global_atomic_add_f32 (STOREcnt, fire-and-forget)
  unsafeAtomicAdd(oi + 0, he);
  unsafeAtomicAdd(oi + 1, fx);
  unsafeAtomicAdd(oi + 2, fy);
  unsafeAtomicAdd(oi + 3, fz);
  unsafeAtomicAdd(oj + 0, he);
  unsafeAtomicAdd(oj + 1, -fx);
  unsafeAtomicAdd(oj + 2, -fy);
  unsafeAtomicAdd(oj + 3, -fz);
}

// Stage one 128-pair tile (1536 B = 384 dwords) of `distances` into LDS with
// three async b128 loads per lane (lane l covers 16 B at offsets 0/512/1024).
// The same INST_OFFSET applies to both the global and the LDS address per the
// ISA, so one (ldsb, voff) pair covers all three instructions. Non-temporal:
// the stream is read exactly once.
__device__ __forceinline__ void stage_tile_async(uint32_t ldsb, uint32_t voff,
                                                 const float* dist) {
  asm volatile(
      "global_load_async_to_lds_b128 %0, %1, %2 offset:0 th:TH_LOAD_NT\n\t"
      "global_load_async_to_lds_b128 %0, %1, %2 offset:512 th:TH_LOAD_NT\n\t"
      "global_load_async_to_lds_b128 %0, %1, %2 offset:1024 th:TH_LOAD_NT"
      :
      : "v"(ldsb), "v"(voff), "s"(dist)
      : "memory");
}

__global__ void __launch_bounds__(256)
lj_main(const float* __restrict__ dist, const int* __restrict__ pi,
        const int* __restrict__ pj, float* __restrict__ out,
        int nTiles, int nWaves) {
  __shared__ float smem[WAVES_PER_BLOCK * 2 * TILE * 3];  // 24 KB / block

  const int lane = threadIdx.x & 31;
  const int wib  = threadIdx.x >> 5;
  const int W    = (blockIdx.x * (int)blockDim.x + (int)threadIdx.x) >> 5;

  float* wbuf = smem + wib * (2 * TILE * 3);
  const uint32_t lds0      = lds_byte_off(wbuf) + (uint32_t)lane * 16u;
  const uint32_t bufStride = (uint32_t)TILE_BYTES;  // 1536 B

  int t = W;
  if (t < nTiles) {
    stage_tile_async(lds0,
                     (uint32_t)t * (uint32_t)TILE_BYTES + (uint32_t)lane * 16u,
                     dist);
  }
  int b = 0;
  for (; t < nTiles; t += nWaves) {
    const int tn = t + nWaves;
    if (tn < nTiles) {
      // issue next tile's async loads into the other buffer, prefetch the
      // upcoming pair-index cachelines, then wait for THIS tile's 3 loads
      // (async loads complete in order: <=3 outstanding => old 3 done).
      stage_tile_async(
          lds0 + (uint32_t)(b ^ 1) * bufStride,
          (uint32_t)tn * (uint32_t)TILE_BYTES + (uint32_t)lane * 16u, dist);
      __builtin_prefetch(pi + tn * TILE + lane * 4, 0, 0);
      __builtin_prefetch(pj + tn * TILE + lane * 4, 0, 0);
      asm volatile("s_wait_asynccnt 3" ::: "memory");
    } else {
      asm volatile("s_wait_asynccnt 0" ::: "memory");
    }
#pragma unroll
    for (int k = 0; k < SUB; ++k) {
      const int p  = t * TILE + k * 32 + lane;
      const int ai = __builtin_nontemporal_load(pi + p);
      const int aj = __builtin_nontemporal_load(pj + p);
      // stride-3 dword LDS read: gcd(3,64)==1 -> bank-conflict-free
      const float* src = wbuf + b * (TILE * 3) + (k * 32 + lane) * 3;
      lj_pair(src[0], src[1], src[2], ai, aj, out);
    }
    b ^= 1;
  }
}

// scalar tail for P % 128 (not hit for P = 8388608)
__global__ void __launch_bounds__(256)
lj_tail(const float* __restrict__ dist, const int* __restrict__ pi,
        const int* __restrict__ pj, float* __restrict__ out,
        int start, int P) {
  const int p = start + blockIdx.x * (int)blockDim.x + (int)threadIdx.x;
  if (p >= P) return;
  lj_pair(dist[3 * p + 0], dist[3 * p + 1], dist[3 * p + 2],
          pi[p], pj[p], out);
}

__global__ void __launch_bounds__(256)
zero4(float4* __restrict__ o, int n4) {
  const int i = blockIdx.x * (int)blockDim.x + (int)threadIdx.x;
  if (i < n4) o[i] = make_float4(0.f, 0.f, 0.f, 0.f);
}

extern "C" void kernel_launch(void* const* d_in, const int* in_sizes, int n_in,
                              void* d_out, int out_size, void* d_ws,
                              size_t ws_size, hipStream_t stream) {
  (void)n_in; (void)d_ws; (void)ws_size;
  const float* dist = (const float*)d_in[0];  // [P,3] f32
  const int*   pi   = (const int*)d_in[1];    // [P] i32
  const int*   pj   = (const int*)d_in[2];    // [P] i32
  float*       out  = (float*)d_out;          // [N,4] f32
  const int P = in_sizes[1];

  // zero the accumulation buffer (harness poisons d_out)
  const int n4 = out_size / 4;
  if (n4 > 0) {
    zero4<<<(n4 + 255) / 256, 256, 0, stream>>>((float4*)out, n4);
  }

  const int nTiles = P / TILE;
  if (nTiles > 0) {
    int nW     = (nTiles + TILES_PER_WAVE - 1) / TILES_PER_WAVE;
    int blocks = (nW + WAVES_PER_BLOCK - 1) / WAVES_PER_BLOCK;
    int nWaves = blocks * WAVES_PER_BLOCK;  // stride must match launch
    lj_main<<<blocks, 256, 0, stream>>>(dist, pi, pj, out, nTiles, nWaves);
  }
  const int rem = P - nTiles * TILE;
  if (rem > 0) {
    lj_tail<<<(rem + 255) / 256, 256, 0, stream>>>(dist, pi, pj, out,
                                                   nTiles * TILE, P);
  }
}